// UltrasoundRendering_10290741641579
// MI455X (gfx1250) — compile-verified
//
#include <hip/hip_runtime.h>

typedef float v2f __attribute__((ext_vector_type(2)));
typedef float v8f __attribute__((ext_vector_type(8)));

#define HH 2048
#define WW 2048
#define NN (2048*2048)

// normalized 1-D Gaussian (size 5, sigma 1): exp(-{4,1,0,1,4}/2)/sum
// branchless: g[d] depends only on |d-2|; zero outside 0<=d<=4
__device__ __forceinline__ float gcoef(int d) {
    int ad = abs(d - 2);
    float v = (ad == 0) ? 0.4026199468942474f
            : ((ad == 1) ? 0.24420134200323332f : 0.05448868454964294f);
    return ((unsigned)d <= 4u) ? v : 0.0f;
}

// branchless zero-padded load from a row/col array
__device__ __forceinline__ float loadz(const float* __restrict__ p, int i, int n) {
    int ic = min(max(i, 0), n - 1);
    float v = p[ic];
    return (i == ic) ? v : 0.0f;
}

__global__ void init_stats_kernel(unsigned* stats) {
    if (threadIdx.x == 0) {
        stats[0] = 0x7F800000u; // +inf  (min slot)
        stats[1] = 0x00000000u; // 0.0   (max slot; attenuation_total > 0)
    }
}

// One block per row: maps, stencils, two row-wise cumprod scans, min/max atomics.
__global__ __launch_bounds__(256) void row_kernel(
    const int* __restrict__ ct,
    const float* __restrict__ dAc, const float* __restrict__ dAt,
    const float* __restrict__ dM0, const float* __restrict__ dM1,
    const float* __restrict__ dS0,
    const float* __restrict__ tex, const float* __restrict__ sp,
    float* __restrict__ bnd, float* __restrict__ rfl, float* __restrict__ sct,
    float* __restrict__ att, float* __restrict__ rto, unsigned* __restrict__ stats)
{
    __shared__ float mAc[14], mAt[14], mM0[14], mM1[14], mS0[14];
    __shared__ float sA[256], sR[256], sMn[256], sMx[256];
    const int t = threadIdx.x;
    const int row = blockIdx.x;
    if (t < 14) { mAc[t] = 0.f; mAt[t] = 0.f; mM0[t] = 0.f; mM1[t] = 0.f; mS0[t] = 0.f; }
    __syncthreads();
    if (t < 10) {
        const int KEYS[10] = {2, 3, 4, 6, 8, 9, 10, 11, 12, 13};
        int k = KEYS[t];
        mAc[k] = dAc[t]; mAt[k] = dAt[t]; mM0[k] = dM0[t]; mM1[k] = dM1[t]; mS0[k] = dS0[t];
    }
    __syncthreads();

    const float dist = 2.0f / 2047.0f;
    const int base = t * 8;
    const size_t rowoff = (size_t)row * WW;
    const int* ctrow = ct + rowoff;

    // vectorized bulk loads (base is 32B aligned)
    const int4*   ct4  = reinterpret_cast<const int4*>(ctrow) + t * 2;
    const float4* tex4 = reinterpret_cast<const float4*>(tex + rowoff) + t * 2;
    const float4* sp4  = reinterpret_cast<const float4*>(sp + rowoff) + t * 2;
    int4   ca = ct4[0],  cb = ct4[1];
    float4 ta = tex4[0], tb = tex4[1];
    float4 pa4 = sp4[0], pb4 = sp4[1];
    // halo ct loads, clamped (OOB values are provably unused by the selects below)
    int ctm1 = ctrow[max(base - 1, 0)];
    int ctp8 = ctrow[min(base + 8, WW - 1)];

    int   ctv[10] = {ctm1, ca.x, ca.y, ca.z, ca.w, cb.x, cb.y, cb.z, cb.w, ctp8};
    float txv[8]  = {ta.x, ta.y, ta.z, ta.w, tb.x, tb.y, tb.z, tb.w};
    float spv[8]  = {pa4.x, pa4.y, pa4.z, pa4.w, pb4.x, pb4.y, pb4.z, pb4.w};

    float Av[10];
    #pragma unroll
    for (int k = 0; k < 10; k++) {
        int c = ctv[k] & 15; c = min(c, 13);
        ctv[k] = c; Av[k] = mAc[c];
    }

    float pa[8], pr[8], bb[8], rr[8], sv[8];
    float runA = 1.0f, runR = 1.0f;
    #pragma unroll
    for (int k = 0; k < 8; k++) {
        int j = base + k;
        int c = ctv[k + 1];
        float Aj = Av[k + 1];
        float diff = (j < WW - 1) ? (Aj - Av[k + 2]) : 0.0f;
        float sum0 = (j > 0) ? (Aj + Av[k]) : Aj;
        float sum = (sum0 == 0.0f) ? 1.0f : sum0;
        float b = 1.0f - __expf(-diff * diff * 10.0f);      // 1/ALPHA_BOUNDARY = 10
        float x = diff / sum;
        float r = 1.0f / (1.0f + __expf(-x * x));
        float af = __expf(-mAt[c] * dist);
        float sg = 1.0f / (1.0f + __expf(-10.0f * (mM0[c] - spv[k])));
        bb[k] = b;
        rr[k] = r;
        sv[k] = (1.0f - sg) * (txv[k] * mS0[c] + mM1[c]);
        runA *= af;                pa[k] = runA;
        runR *= (1.0f - r * b);    pr[k] = runR;
    }
    // vectorized map stores
    {
        float4* b4 = reinterpret_cast<float4*>(bnd + rowoff) + t * 2;
        float4* r4 = reinterpret_cast<float4*>(rfl + rowoff) + t * 2;
        float4* s4 = reinterpret_cast<float4*>(sct + rowoff) + t * 2;
        b4[0] = make_float4(bb[0], bb[1], bb[2], bb[3]);
        b4[1] = make_float4(bb[4], bb[5], bb[6], bb[7]);
        r4[0] = make_float4(rr[0], rr[1], rr[2], rr[3]);
        r4[1] = make_float4(rr[4], rr[5], rr[6], rr[7]);
        s4[0] = make_float4(sv[0], sv[1], sv[2], sv[3]);
        s4[1] = make_float4(sv[4], sv[5], sv[6], sv[7]);
    }

    sA[t] = runA; sR[t] = runR;
    __syncthreads();
    // inclusive multiplicative scan over 256 thread totals
    #pragma unroll
    for (int off = 1; off < 256; off <<= 1) {
        float va = (t >= off) ? sA[t - off] : 1.0f;
        float vr = (t >= off) ? sR[t - off] : 1.0f;
        __syncthreads();
        sA[t] *= va; sR[t] *= vr;
        __syncthreads();
    }
    float prefA = (t > 0) ? sA[t - 1] : 1.0f;
    float prefR = (t > 0) ? sR[t - 1] : 1.0f;

    float av[8], rv[8];
    float mn = 3.4e38f, mx = -3.4e38f;
    #pragma unroll
    for (int k = 0; k < 8; k++) {
        av[k] = prefA * pa[k];
        rv[k] = prefR * pr[k];
        mn = fminf(mn, av[k]); mx = fmaxf(mx, av[k]);
    }
    {
        float4* a4 = reinterpret_cast<float4*>(att + rowoff) + t * 2;
        float4* o4 = reinterpret_cast<float4*>(rto + rowoff) + t * 2;
        a4[0] = make_float4(av[0], av[1], av[2], av[3]);
        a4[1] = make_float4(av[4], av[5], av[6], av[7]);
        o4[0] = make_float4(rv[0], rv[1], rv[2], rv[3]);
        o4[1] = make_float4(rv[4], rv[5], rv[6], rv[7]);
    }

    sMn[t] = mn; sMx[t] = mx;
    __syncthreads();
    for (int off = 128; off > 0; off >>= 1) {
        if (t < off) { sMn[t] = fminf(sMn[t], sMn[t + off]); sMx[t] = fmaxf(sMx[t], sMx[t + off]); }
        __syncthreads();
    }
    if (t == 0) {
        // attenuation_total is strictly positive -> uint bit order == float order
        atomicMin(&stats[0], __float_as_uint(sMn[0]));
        atomicMax(&stats[1], __float_as_uint(sMx[0]));
    }
}

// Horizontal (axis-1) 5-tap Gaussian pass as banded matmul: out_h = X(16x20) * T(20x16),
// T[p][q] = g[p-q] for 0<=p-q<=4, decomposed into 5 K=4 WMMA steps. Two inputs at once.
__global__ __launch_bounds__(32) void convH_kernel(
    const float* __restrict__ sct, const float* __restrict__ bnd,
    float* __restrict__ sh, float* __restrict__ bh)
{
    const int l = threadIdx.x;
    const int half = l >> 4;   // A: k-pair select; B: row-pair select
    const int q = l & 15;      // A: matrix row m; B: matrix col n
    const int ti = blockIdx.y, tj = blockIdx.x;
    const int gr = ti * 16 + q;
    const float* srow = sct + (size_t)gr * WW;
    const float* brow = bnd + (size_t)gr * WW;

    v8f accS = {}; v8f accB = {};
    #pragma unroll
    for (int c = 0; c < 5; c++) {
        int p0 = 4 * c + 2 * half;          // banded index of this lane's first K element
        int col0 = tj * 16 - 2 + p0;        // global input column for A reg 0
        v2f aS, aB, b;
        aS.x = loadz(srow, col0, WW);
        aS.y = loadz(srow, col0 + 1, WW);
        aB.x = loadz(brow, col0, WW);
        aB.y = loadz(brow, col0 + 1, WW);
        b.x = gcoef(p0 - q);
        b.y = gcoef(p0 + 1 - q);
        accS = __builtin_amdgcn_wmma_f32_16x16x4_f32(false, aS, false, b, (short)0, accS, false, false);
        accB = __builtin_amdgcn_wmma_f32_16x16x4_f32(false, aB, false, b, (short)0, accB, false, false);
    }
    #pragma unroll
    for (int r = 0; r < 8; r++) {
        int orow = ti * 16 + r + 8 * half;  // C/D layout: VGPR r -> M=r (lo lanes) / r+8 (hi lanes)
        int ocol = tj * 16 + q;
        sh[(size_t)orow * WW + ocol] = accS[r];
        bh[(size_t)orow * WW + ocol] = accB[r];
    }
}

// Vertical pass (A = constant 16x20 band, B = data 20x16) + fused normalization/combine/output.
__global__ __launch_bounds__(32) void convV_kernel(
    const float* __restrict__ sh, const float* __restrict__ bh,
    const float* __restrict__ att, const float* __restrict__ rto,
    const float* __restrict__ rfl, const float* __restrict__ mask,
    const unsigned* __restrict__ stats, float* __restrict__ out)
{
    const int l = threadIdx.x;
    const int half = l >> 4;
    const int q = l & 15;
    const int ti = blockIdx.y, tj = blockIdx.x;
    const int col = tj * 16 + q;

    __builtin_prefetch(att + (size_t)(ti * 16) * WW + col, 0, 0);

    v8f accS = {}; v8f accB = {};
    #pragma unroll
    for (int c = 0; c < 5; c++) {
        int p0 = 4 * c + 2 * half;
        v2f a, bS, bB;
        a.x = gcoef(p0 - q);        // A[m][k] = Tc[m][p] = g[p-m], lane m = q
        a.y = gcoef(p0 + 1 - q);
        int r0 = ti * 16 - 2 + p0;  // global input row for B reg 0
        // branchless row-clamped loads, zeroed when out of range
        int r0c = min(max(r0, 0), HH - 1);
        int r1c = min(max(r0 + 1, 0), HH - 1);
        float s0 = sh[(size_t)r0c * WW + col];
        float s1 = sh[(size_t)r1c * WW + col];
        float b0 = bh[(size_t)r0c * WW + col];
        float b1 = bh[(size_t)r1c * WW + col];
        bS.x = (r0 == r0c) ? s0 : 0.0f;
        bS.y = (r0 + 1 == r1c) ? s1 : 0.0f;
        bB.x = (r0 == r0c) ? b0 : 0.0f;
        bB.y = (r0 + 1 == r1c) ? b1 : 0.0f;
        accS = __builtin_amdgcn_wmma_f32_16x16x4_f32(false, a, false, bS, (short)0, accS, false, false);
        accB = __builtin_amdgcn_wmma_f32_16x16x4_f32(false, a, false, bB, (short)0, accB, false, false);
    }

    float mn = __uint_as_float(stats[0]);
    float mx = __uint_as_float(stats[1]);
    float scale = 1.0f / (mx - mn);

    // us_mask.T column for this lane: 8 consecutive rows -> two b128 loads
    const float4* m4 = reinterpret_cast<const float4*>(mask + (size_t)col * WW + ti * 16 + 8 * half);
    float4 m0 = m4[0], m1 = m4[1];
    float mv[8] = {m0.x, m0.y, m0.z, m0.w, m1.x, m1.y, m1.z, m1.w};

    #pragma unroll
    for (int r = 0; r < 8; r++) {
        int row = ti * 16 + r + 8 * half;
        size_t idx = (size_t)row * WW + col;
        float at = (att[idx] - mn) * scale;
        float bv = at * accS[r];
        float rv = at * rto[idx] * rfl[idx] * accB[r];
        float inten = fminf(fmaxf(bv + rv, 0.0f), 1.0f);
        out[idx] = inten * mv[r];   // us_mask.T
    }
}

extern "C" void kernel_launch(void* const* d_in, const int* in_sizes, int n_in,
                              void* d_out, int out_size, void* d_ws, size_t ws_size,
                              hipStream_t stream)
{
    const int*   ct   = (const int*)d_in[0];
    const float* dAc  = (const float*)d_in[1];
    const float* dAt  = (const float*)d_in[2];
    const float* dM0  = (const float*)d_in[3];
    const float* dM1  = (const float*)d_in[4];
    const float* dS0  = (const float*)d_in[5];
    const float* mask = (const float*)d_in[6];
    const float* tex  = (const float*)d_in[7];
    const float* sp   = (const float*)d_in[8];
    float* out = (float*)d_out;

    float* Wp  = (float*)d_ws;
    float* bnd = Wp;
    float* rfl = Wp + 1 * (size_t)NN;
    float* sct = Wp + 2 * (size_t)NN;
    float* att = Wp + 3 * (size_t)NN;
    float* rto = Wp + 4 * (size_t)NN;
    float* sh  = Wp + 5 * (size_t)NN;
    float* bh  = Wp + 6 * (size_t)NN;
    unsigned* stats = (unsigned*)(Wp + 7 * (size_t)NN);

    hipLaunchKernelGGL(init_stats_kernel, dim3(1), dim3(32), 0, stream, stats);
    hipLaunchKernelGGL(row_kernel, dim3(HH), dim3(256), 0, stream,
                       ct, dAc, dAt, dM0, dM1, dS0, tex, sp,
                       bnd, rfl, sct, att, rto, stats);
    hipLaunchKernelGGL(convH_kernel, dim3(128, 128), dim3(32), 0, stream, sct, bnd, sh, bh);
    hipLaunchKernelGGL(convV_kernel, dim3(128, 128), dim3(32), 0, stream,
                       sh, bh, att, rto, rfl, mask, stats, out);
}